// ChannelWiseLayerNorm_40999757807760
// MI455X (gfx1250) — compile-verified
//
#include <hip/hip_runtime.h>
#include <math.h>

typedef __attribute__((ext_vector_type(2))) float v2f;
typedef __attribute__((ext_vector_type(8))) float v8f;

#define EPS 1e-8f

// ---------------------------------------------------------------------------
// Pass 1: per-(b,t) channel reduction via V_WMMA_F32_16X16X4_F32.
// A = all-ones 16x4  =>  D[m][n] = sum_k B[k][n]  for every row m.
// Each wave owns a 16-wide t tile; loop over C in chunks of 4 channels.
// Lanes 0-15 carry K rows {khalf, khalf+1} = {0,1}; lanes 16-31 carry {2,3}.
// Because A is all-ones the K->VGPR assignment cannot affect the result
// (summation over K is permutation invariant); only the documented C/D
// column mapping (N = lane&15) matters.
// ---------------------------------------------------------------------------
__global__ void __launch_bounds__(256)
stats_kernel(const float* __restrict__ x,
             float* __restrict__ step_sum,
             float* __restrict__ step_pow,
             int C, int T) {
    const int b    = blockIdx.y;
    const int wave = threadIdx.x >> 5;
    const int lane = threadIdx.x & 31;
    const int t0   = blockIdx.x * 128 + wave * 16;   // 8 waves * 16 t = 128 t/block
    const int n    = lane & 15;                      // D column = t0 + n
    const int kh   = (lane >> 4) << 1;               // 0 or 2

    const float* p = x + ((size_t)b * C + kh) * (size_t)T + (size_t)(t0 + n);
    const size_t cstride = 4 * (size_t)T;

    v2f a; a[0] = 1.0f; a[1] = 1.0f;                 // all-ones A matrix
    v8f c_sum = {};
    v8f c_sq  = {};

    for (int c0 = 0; c0 < C; c0 += 4) {
        __builtin_prefetch((const void*)(p + 2 * cstride), 0, 1); // global_prefetch_b8
        float b0 = p[0];
        float b1 = p[T];
        v2f bv; bv[0] = b0;      bv[1] = b1;
        v2f bq; bq[0] = b0 * b0; bq[1] = b1 * b1;
        c_sum = __builtin_amdgcn_wmma_f32_16x16x4_f32(false, a, false, bv,
                                                      (short)0, c_sum, false, false);
        c_sq  = __builtin_amdgcn_wmma_f32_16x16x4_f32(false, a, false, bq,
                                                      (short)0, c_sq,  false, false);
        p += cstride;
    }

    // D layout: VGPR0, lanes 0-15 hold row M=0, N=lane  => the channel sum for t0+lane
    if (lane < 16) {
        step_sum[(size_t)b * T + t0 + lane] = c_sum[0];
        step_pow[(size_t)b * T + t0 + lane] = c_sq[0];
    }
}

// ---------------------------------------------------------------------------
// Pass 2: per-batch inclusive prefix scan over T of (step_sum, step_pow_sum),
// then convert in place to (cum_mean, 1/cum_std) using the reference's exact
// variance formula. One block per batch; wave32 shuffle scan + LDS combine.
// ---------------------------------------------------------------------------
__global__ void __launch_bounds__(1024)
scan_kernel(float* __restrict__ step_sum,
            float* __restrict__ step_pow,
            int C, int T) {
    const int b = blockIdx.x;
    float* ss = step_sum + (size_t)b * T;
    float* sp = step_pow + (size_t)b * T;

    const int tid    = threadIdx.x;
    const int lane   = tid & 31;
    const int wave   = tid >> 5;
    const int nwaves = blockDim.x >> 5;

    __shared__ float s_wa[32];
    __shared__ float s_wb[32];

    float carry_a = 0.0f, carry_b = 0.0f;
    const float Cf = (float)C;

    for (int base = 0; base < T; base += blockDim.x) {
        const int t = base + tid;
        float a = 0.0f, q = 0.0f;
        if (t < T) { a = ss[t]; q = sp[t]; }

        // wave32 inclusive scan (shuffle-based, no barriers)
        float as = a, qs = q;
#pragma unroll
        for (int off = 1; off < 32; off <<= 1) {
            float ta = __shfl_up(as, off, 32);
            float tq = __shfl_up(qs, off, 32);
            if (lane >= off) { as += ta; qs += tq; }
        }
        if (lane == 31) { s_wa[wave] = as; s_wb[wave] = qs; }
        __syncthreads();

        float woff_a = 0.0f, woff_b = 0.0f;
        float tot_a  = 0.0f, tot_b  = 0.0f;
        for (int w = 0; w < nwaves; ++w) {
            float va = s_wa[w], vb = s_wb[w];
            if (w < wave) { woff_a += va; woff_b += vb; }
            tot_a += va; tot_b += vb;
        }

        const float cum_a = carry_a + woff_a + as;   // cum_sum
        const float cum_b = carry_b + woff_b + qs;   // cum_pow_sum

        if (t < T) {
            const float cnt  = (float)(t + 1) * Cf;
            const float mean = cum_a / cnt;
            const float var  = (cum_b - 2.0f * mean * cum_a) / cnt + mean * mean;
            const float istd = 1.0f / sqrtf(var + EPS);
            ss[t] = mean;   // overwrite in place
            sp[t] = istd;
        }

        carry_a += tot_a;
        carry_b += tot_b;
        __syncthreads();   // protect s_wa/s_wb before next chunk
    }
}

// ---------------------------------------------------------------------------
// Pass 3: streaming normalize, float4 along T. mean/istd arrays are ~1 MB,
// L2-resident, reused by all 512 channels.
// ---------------------------------------------------------------------------
__global__ void __launch_bounds__(256)
norm_kernel(const float* __restrict__ x,
            const float* __restrict__ gain,
            const float* __restrict__ bias,
            const float* __restrict__ mean_arr,
            const float* __restrict__ istd_arr,
            float* __restrict__ out,
            int C, int T) {
    const int b  = blockIdx.z;
    const int c  = blockIdx.y;
    const int t4 = blockIdx.x * blockDim.x + threadIdx.x;
    const int t  = t4 * 4;
    if (t >= T) return;

    const size_t row = ((size_t)b * C + c) * (size_t)T + (size_t)t;
    const size_t bt  = (size_t)b * T + (size_t)t;

    const float4 xv = *reinterpret_cast<const float4*>(x + row);
    const float4 mv = *reinterpret_cast<const float4*>(mean_arr + bt);
    const float4 sv = *reinterpret_cast<const float4*>(istd_arr + bt);
    const float g  = gain[c];
    const float bi = bias[c];

    float4 o;
    o.x = (xv.x - mv.x) * sv.x * g + bi;
    o.y = (xv.y - mv.y) * sv.y * g + bi;
    o.z = (xv.z - mv.z) * sv.z * g + bi;
    o.w = (xv.w - mv.w) * sv.w * g + bi;
    *reinterpret_cast<float4*>(out + row) = o;
}

extern "C" void kernel_launch(void* const* d_in, const int* in_sizes, int n_in,
                              void* d_out, int out_size, void* d_ws, size_t ws_size,
                              hipStream_t stream) {
    (void)n_in; (void)out_size; (void)ws_size;
    const float* x    = (const float*)d_in[0];
    const float* gain = (const float*)d_in[1];
    const float* bias = (const float*)d_in[2];
    float* out = (float*)d_out;

    const int C = in_sizes[1];                 // 512
    const int T = 16000;
    const int B = in_sizes[0] / (C * T);       // 8

    float* step_sum = (float*)d_ws;            // [B*T] -> becomes cum_mean
    float* step_pow = step_sum + (size_t)B * T; // [B*T] -> becomes 1/cum_std

    // Pass 1: channel reductions via WMMA (grid 125 x 8, exact tiling)
    dim3 g1(T / 128, B);
    stats_kernel<<<g1, 256, 0, stream>>>(x, step_sum, step_pow, C, T);

    // Pass 2: causal scan + mean/istd (one block per batch)
    scan_kernel<<<B, 1024, 0, stream>>>(step_sum, step_pow, C, T);

    // Pass 3: normalize
    dim3 g3((T / 4 + 255) / 256, C, B);
    norm_kernel<<<g3, 256, 0, stream>>>(x, gain, bias, step_sum, step_pow, out, C, T);
}